// GaussianSplat3D_90151363543811
// MI455X (gfx1250) — compile-verified
//
#include <hip/hip_runtime.h>

typedef __attribute__((ext_vector_type(2))) float v2f;
typedef __attribute__((ext_vector_type(8))) float v8f;

#define NGAUSS 4096
#define CH 128
#define IMG_W 512
#define IMG_H 512

// ---------------------------------------------------------------------------
// Phase 1: per-gaussian projection, 2D inverse covariance, sort keys
// ---------------------------------------------------------------------------
__global__ __launch_bounds__(256) void gs_preprocess(
    const float* __restrict__ means, const float* __restrict__ log_scales,
    const float* __restrict__ colors, const float* __restrict__ opacities,
    const float* __restrict__ intr, const float* __restrict__ c2w,
    float4* __restrict__ featA, float4* __restrict__ featB,
    float* __restrict__ featC, unsigned long long* __restrict__ keys, int N)
{
  int n = blockIdx.x * blockDim.x + threadIdx.x;
  if (n >= N) return;

  float fx = intr[0], cx = intr[2], fy = intr[4], cy = intr[5];
  float R00=c2w[0], R01=c2w[1], R02=c2w[2],  tx=c2w[3];
  float R10=c2w[4], R11=c2w[5], R12=c2w[6],  ty=c2w[7];
  float R20=c2w[8], R21=c2w[9], R22=c2w[10], tz=c2w[11];

  float d0 = means[n*3+0]-tx, d1 = means[n*3+1]-ty, d2 = means[n*3+2]-tz;
  // cam = (m - t) @ R   =>  cam[j] = sum_i d_i * R[i][j]
  float x = d0*R00 + d1*R10 + d2*R20;
  float y = d0*R01 + d1*R11 + d2*R21;
  float z = d0*R02 + d1*R12 + d2*R22;

  bool valid = z > 1e-4f;
  float zs = fmaxf(z, 1e-4f);
  float u = fx*x/zs + cx;
  float v = fy*y/zs + cy;

  float s0 = __expf(2.f*log_scales[n*3+0]) + 1e-6f;
  float s1 = __expf(2.f*log_scales[n*3+1]) + 1e-6f;
  float s2 = __expf(2.f*log_scales[n*3+2]) + 1e-6f;

  // M = R^T diag(s) R (symmetric):  M[i][k] = sum_j R[j][i]*s_j*R[j][k]
  float M00 = R00*s0*R00 + R10*s1*R10 + R20*s2*R20;
  float M01 = R00*s0*R01 + R10*s1*R11 + R20*s2*R21;
  float M02 = R00*s0*R02 + R10*s1*R12 + R20*s2*R22;
  float M11 = R01*s0*R01 + R11*s1*R11 + R21*s2*R21;
  float M12 = R01*s0*R02 + R11*s1*R12 + R21*s2*R22;
  float M22 = R02*s0*R02 + R12*s1*R12 + R22*s2*R22;

  float izs = 1.f/zs;
  float j00 = fx*izs, j02 = -fx*x*izs*izs;
  float j11 = fy*izs, j12 = -fy*y*izs*izs;

  // cov2 = J M J^T
  float t0x = M00*j00 + M02*j02;
  float t0z = M02*j00 + M22*j02;
  float t1x = M01*j11 + M02*j12;
  float t1y = M11*j11 + M12*j12;
  float t1z = M12*j11 + M22*j12;
  float a = j00*t0x + j02*t0z + 1e-4f;
  float b = j00*t1x + j02*t1z;
  float c = j11*t1y + j12*t1z + 1e-4f;

  float det  = fmaxf(a*c - b*b, 1e-12f);
  float idet = 1.f/det;
  float ia = c*idet, ib = -b*idet, ic = a*idet;

  float op = (1.f/(1.f + __expf(-opacities[n]))) * (valid ? 1.f : 0.f);
  float cr = fminf(fmaxf(colors[n*3+0],0.f),1.f);
  float cg = fminf(fmaxf(colors[n*3+1],0.f),1.f);
  float cb = fminf(fmaxf(colors[n*3+2],0.f),1.f);

  featA[n] = make_float4(u, v, ia, ib);
  featB[n] = make_float4(ic, op, cr, cg);
  featC[n] = cb;

  // stable depth key: positive float bits are order-preserving as uint
  unsigned zb = valid ? __float_as_uint(z) : 0x7F800000u; // +inf if invalid
  keys[n] = ((unsigned long long)zb << 32) | (unsigned)n;
}

// ---------------------------------------------------------------------------
// Phase 2: single-workgroup bitonic sort of 4096 (depth,index) keys in LDS
// ---------------------------------------------------------------------------
__global__ __launch_bounds__(1024) void gs_sort(
    const unsigned long long* __restrict__ keysIn, unsigned* __restrict__ order)
{
  __shared__ unsigned long long k[NGAUSS]; // 32 KB of 320 KB/WGP
  int tid = threadIdx.x;
  for (int i = tid; i < NGAUSS; i += 1024) k[i] = keysIn[i];

  for (int size = 2; size <= NGAUSS; size <<= 1) {
    for (int stride = size >> 1; stride > 0; stride >>= 1) {
      __syncthreads();
      for (int t = tid; t < NGAUSS; t += 1024) {
        int ixj = t ^ stride;
        if (ixj > t) {
          bool ascend = ((t & size) == 0);
          unsigned long long va = k[t], vb = k[ixj];
          if (ascend ? (va > vb) : (va < vb)) { k[t] = vb; k[ixj] = va; }
        }
      }
    }
  }
  __syncthreads();
  for (int i = tid; i < NGAUSS; i += 1024)
    order[i] = (unsigned)(k[i] & 0xFFFFFFFFu);
}

// ---------------------------------------------------------------------------
// Phase 3: tiled front-to-back compositing.
// Block = 256 threads = 8 waves; each wave owns a 16-pixel row segment.
// Lane = m + 16*h: pixel m, gaussian sub-pair h. Per group of 4 gaussians:
// each lane computes 2 alphas; __shfl_xor(16) passes the (1-a0)(1-a1)
// prefix so weights are exact front-to-back; V_WMMA_F32_16X16X4_F32
// accumulates weights(16x4) x colors(4x16, 3 used cols) into a 16x16 C.
// Colors are staged TRANSPOSED (sColT[4][CH], row 3 zeroed) so the B-matrix
// fetch is one unconditional aligned float2 LDS load — no EXEC divergence.
// ---------------------------------------------------------------------------
__global__ __launch_bounds__(256) void gs_render(
    const float4* __restrict__ featA, const float4* __restrict__ featB,
    const float* __restrict__ featC, const unsigned* __restrict__ order,
    float* __restrict__ img, int N)
{
  __shared__ float4 sA[CH];          // u, v, ia, ib
  __shared__ float2 sB[CH];          // ic, op
  __shared__ float  sColT[4 * CH];   // rows 0..2 = R,G,B; row 3 = zeros

  const int tid  = threadIdx.x;
  const int wave = tid >> 5;
  const int lane = tid & 31;
  const int m = lane & 15;   // pixel within wave's strip / N column
  const int h = lane >> 4;   // which gaussian pair of the group of 4
  const int colBase = ((m < 3) ? m : 3) * CH;   // per-lane constant

  const int tileX = blockIdx.x;                 // 32 tiles of 16 px
  const int rowY  = blockIdx.y * 8 + wave;      // 64 blocks * 8 waves
  const float pxf = (float)(tileX * 16 + m) + 0.5f;
  const float pyf = (float)rowY + 0.5f;

  float T = 1.0f;     // transmittance (replicated across the lane pair)
  v8f acc = {};       // C/D accumulator: 16 pixels x 16 channels (3 used)

  for (int cb = 0; cb < N; cb += CH) {
    __syncthreads();
    if (tid < CH) {
      unsigned si = order[cb + tid];
      float4 fa = featA[si];
      float4 fb = featB[si];
      float  fc = featC[si];
      sA[tid] = fa;
      sB[tid] = make_float2(fb.x, fb.y);
      sColT[0 * CH + tid] = fb.z;
      sColT[1 * CH + tid] = fb.w;
      sColT[2 * CH + tid] = fc;
      sColT[3 * CH + tid] = 0.0f;
      if (cb + CH < N) __builtin_prefetch(&order[cb + CH + tid], 0, 0);
    }
    __syncthreads();

#pragma unroll 2
    for (int g = 0; g < CH; g += 4) {
      const int g0 = g + 2 * h;            // even -> 8B-aligned float2 loads
      float4 fa0 = sA[g0];
      float4 fa1 = sA[g0 + 1];
      float2 fb0 = sB[g0];
      float2 fb1 = sB[g0 + 1];
      float2 col = *(const float2*)&sColT[colBase + g0];

      float dx0 = pxf - fa0.x, dy0 = pyf - fa0.y;
      float q0  = fa0.z*dx0*dx0 + 2.f*fa0.w*dx0*dy0 + fb0.x*dy0*dy0;
      float w0  = (q0 <= 9.0f) ? __expf(-0.5f*q0) : 0.0f;
      float al0 = fminf(fb0.y * w0, 0.999f);

      float dx1 = pxf - fa1.x, dy1 = pyf - fa1.y;
      float q1  = fa1.z*dx1*dx1 + 2.f*fa1.w*dx1*dy1 + fb1.x*dy1*dy1;
      float w1  = (q1 <= 9.0f) ? __expf(-0.5f*q1) : 0.0f;
      float al1 = fminf(fb1.y * w1, 0.999f);

      float tl      = (1.f - al0) * (1.f - al1);   // this pair's (1-a) product
      float tlOther = __shfl_xor(tl, 16, 32);      // partner pair's product
      float pre     = h ? tlOther : 1.0f;          // front-to-back prefix

      // A-matrix 16x4 F32: lane m+16h holds (M=m,K=2h) and (M=m,K=2h+1)
      v2f A; A.x = al0 * pre * T;
             A.y = al1 * (1.f - al0) * pre * T;

      // B-matrix 4x16 F32: lane n+16h holds (K=2h,N=n) and (K=2h+1,N=n)
      v2f B; B.x = col.x;
             B.y = col.y;

      acc = __builtin_amdgcn_wmma_f32_16x16x4_f32(
          false, A, false, B, (short)0, acc, false, false);

      T *= tl * tlOther;
    }
  }

  // D layout: VGPR r -> (M=r, N=lane) for lanes 0-15, (M=r+8, N=lane-16) for 16-31.
  // Lanes with channel (lane&15) < 3 store 8 pixels each: 6 lanes x 8 = 48 floats.
  if (m < 3) {
#pragma unroll
    for (int r = 0; r < 8; ++r) {
      int px = tileX * 16 + (r + 8 * h);
      img[(rowY * IMG_W + px) * 3 + m] = acc[r];
    }
  }
}

// ---------------------------------------------------------------------------
extern "C" void kernel_launch(void* const* d_in, const int* in_sizes, int n_in,
                              void* d_out, int out_size, void* d_ws, size_t ws_size,
                              hipStream_t stream) {
  const float* means      = (const float*)d_in[0];
  const float* log_scales = (const float*)d_in[1];
  const float* colors     = (const float*)d_in[2];
  const float* opacities  = (const float*)d_in[3];
  const float* intr       = (const float*)d_in[4];
  const float* c2w        = (const float*)d_in[5];
  const int N = in_sizes[0] / 3; // 4096

  // workspace layout
  float4* featA = (float4*)d_ws;                       // N*16 B
  float4* featB = featA + N;                           // N*16 B
  float*  featC = (float*)(featB + N);                 // N*4  B
  unsigned long long* keys = (unsigned long long*)(featC + N); // N*8 B (8B aligned)
  unsigned* order = (unsigned*)(keys + N);             // N*4  B

  float* img = (float*)d_out; // 512*512*3 f32, fully overwritten

  gs_preprocess<<<(N + 255) / 256, 256, 0, stream>>>(
      means, log_scales, colors, opacities, intr, c2w,
      featA, featB, featC, keys, N);

  gs_sort<<<1, 1024, 0, stream>>>(keys, order);

  dim3 grid(IMG_W / 16, IMG_H / 8);
  gs_render<<<grid, 256, 0, stream>>>(featA, featB, featC, order, img, N);
}